// MatchingPursuit_37194416783916
// MI455X (gfx1250) — compile-verified
//
#include <hip/hip_runtime.h>
#include <math.h>

// ---------------------------------------------------------------------------
// ADMM lasso (basis pursuit) for MI455X / gfx1250.  batch=2048, in=1024,
// out=2048, fp32, 100 ADMM iterations + Newton-Schulz inverse of (WW^T + I).
//
// All GEMMs are C[M,N] = A[M,K] @ B^T with B stored [N][K] (k-contiguous):
//   - W is already [out][in] = [N][K] for the W^T products
//   - M, Newton iterates X_k, and M@X_k are symmetric  =>  B == B^T
//   - W is explicitly transposed once for the final decode
// This makes A and B LDS tiles identical ([row][k], stride 36 dwords =>
// 16B-aligned rows), fragments single ds_load_b64, and lets both tiles be
// filled with global_load_async_to_lds_b128 (ASYNCcnt double buffering).
// ---------------------------------------------------------------------------

typedef __attribute__((ext_vector_type(2))) float v2f;
typedef __attribute__((ext_vector_type(8))) float v8f;
typedef int v4i __attribute__((__vector_size__(4 * sizeof(int))));

#define TB   128   // block tile (M and N)
#define TK   32    // K panel
#define LDSP 36    // padded row stride in dwords (mult of 4 -> b128-aligned)

#if __has_builtin(__builtin_amdgcn_global_load_async_to_lds_b128)
#define USE_ASYNC 1
#else
#define USE_ASYNC 0
#endif

#define GLOBAL_AS __attribute__((address_space(1)))
#define LDS_AS    __attribute__((address_space(3)))

__device__ __forceinline__ void copy16_g2l(const float* g, float* l) {
#if USE_ASYNC
  __builtin_amdgcn_global_load_async_to_lds_b128((GLOBAL_AS v4i*)g,
                                                 (LDS_AS v4i*)l, 0, 0);
#else
  *(float4*)l = *(const float4*)g;  // sync fallback
#endif
}

__device__ __forceinline__ void async_wait0() {
#if USE_ASYNC
#if __has_builtin(__builtin_amdgcn_s_wait_asynccnt)
  __builtin_amdgcn_s_wait_asynccnt(0);
#else
  asm volatile("s_wait_asynccnt 0" ::: "memory");
#endif
#endif
}

enum { EPI_NONE = 0, EPI_ADD_IDENT = 1, EPI_2E_MINUS = 2, EPI_ADMM = 3 };

// C[M,N] = A[M,K] @ B^T,  B stored row-major [N][K].
// EPI_ADD_IDENT : C = acc + I
// EPI_2E_MINUS  : C = 2*E - acc                       (Newton-Schulz)
// EPI_ADMM      : zu = acc + U; C = soft(zu,l); U = zu - C
template <int EPI>
__global__ __launch_bounds__(128) void gemm_nt_wmma(
    const float* __restrict__ A, const float* __restrict__ B,
    const float* __restrict__ E, float* __restrict__ C,
    float* __restrict__ U, int M, int N, int K, float lambd) {
  __shared__ float As[2][TB][LDSP];
  __shared__ float Bs[2][TB][LDSP];

  const int tid  = threadIdx.x;   // 128 threads = 4 waves
  const int lane = tid & 31;
  const int wave = tid >> 5;      // 2 (M) x 2 (N) waves, each 64x64
  const int half = lane >> 4;     // K-pair select (ISA 16x4 f32 layout)
  const int lr   = lane & 15;
  const int wm0  = (wave >> 1) * 64;
  const int wn0  = (wave & 1) * 64;
  const int bm   = blockIdx.y * TB;
  const int bn   = blockIdx.x * TB;

  // staging map: 8 passes x (16 rows x 8 lanes-of-float4) covers 128x32
  const int srow = tid >> 3;        // 0..15
  const int scol = (tid & 7) << 2;  // 0,4,..,28

  v8f acc[4][4] = {};

  const int P = K / TK;

  // ---- prologue: panel 0 into buffer 0 ----
#pragma unroll
  for (int p = 0; p < 8; ++p) {
    const int row = srow + p * 16;
    copy16_g2l(A + (size_t)(bm + row) * K + scol, &As[0][row][scol]);
    copy16_g2l(B + (size_t)(bn + row) * K + scol, &Bs[0][row][scol]);
  }
  async_wait0();
  __syncthreads();

  for (int pan = 0; pan < P; ++pan) {
    const int cb = pan & 1;
    // ---- issue next panel into the other buffer (overlaps compute) ----
    if (pan + 1 < P) {
      const int nb = cb ^ 1;
      const int k0 = (pan + 1) * TK;
#pragma unroll
      for (int p = 0; p < 8; ++p) {
        const int row = srow + p * 16;
        copy16_g2l(A + (size_t)(bm + row) * K + k0 + scol, &As[nb][row][scol]);
        copy16_g2l(B + (size_t)(bn + row) * K + k0 + scol, &Bs[nb][row][scol]);
      }
    }
    // ---- compute on current buffer: 8 K-steps x 16 WMMAs ----
#pragma unroll
    for (int kk = 0; kk < TK; kk += 4) {
      v2f af[4], bf[4];
#pragma unroll
      for (int i = 0; i < 4; ++i) {
        af[i] = *(const v2f*)&As[cb][wm0 + i * 16 + lr][kk + 2 * half];
        bf[i] = *(const v2f*)&Bs[cb][wn0 + i * 16 + lr][kk + 2 * half];
      }
#pragma unroll
      for (int i = 0; i < 4; ++i)
#pragma unroll
        for (int j = 0; j < 4; ++j)
          acc[i][j] = __builtin_amdgcn_wmma_f32_16x16x4_f32(
              false, af[i], false, bf[j], (short)0, acc[i][j], false, false);
    }
    async_wait0();
    __syncthreads();
  }

  // ---- epilogue (C/D layout: VGPR g -> row g + 8*half, col = lr) ----
#pragma unroll
  for (int i = 0; i < 4; ++i) {
#pragma unroll
    for (int j = 0; j < 4; ++j) {
      const int col = bn + wn0 + j * 16 + lr;
#pragma unroll
      for (int g = 0; g < 8; ++g) {
        const int row   = bm + wm0 + i * 16 + half * 8 + g;
        const size_t ix = (size_t)row * N + col;
        float val = acc[i][j][g];
        if (EPI == EPI_ADD_IDENT) {
          if (row == col) val += 1.0f;
          C[ix] = val;
        } else if (EPI == EPI_2E_MINUS) {
          C[ix] = 2.0f * E[ix] - val;
        } else if (EPI == EPI_ADMM) {
          const float zu = val + U[ix];
          const float s  = fabsf(zu) - lambd;
          const float zn = (s > 0.0f) ? copysignf(s, zu) : 0.0f;
          C[ix] = zn;       // z (soft threshold)
          U[ix] = zu - zn;  // dual
        } else {
          C[ix] = val;
        }
      }
    }
  }
}

// ------------------------------ elementwise --------------------------------

__global__ void k_init_identity(float* __restrict__ X, int N, float alpha) {
  const size_t idx = (size_t)blockIdx.x * blockDim.x + threadIdx.x;
  const int r = (int)(idx / (size_t)N);
  const int c = (int)(idx % (size_t)N);
  X[idx] = (r == c) ? alpha : 0.0f;
}

__global__ void k_zero2(float4* __restrict__ a, float4* __restrict__ b) {
  const size_t i = (size_t)blockIdx.x * blockDim.x + threadIdx.x;
  const float4 z = {0.0f, 0.0f, 0.0f, 0.0f};
  a[i] = z;
  b[i] = z;
}

// T = Atb + z - u
__global__ void k_residual(float4* __restrict__ T, const float4* __restrict__ Atb,
                           const float4* __restrict__ z,
                           const float4* __restrict__ u) {
  const size_t i = (size_t)blockIdx.x * blockDim.x + threadIdx.x;
  const float4 a = Atb[i], zz = z[i], uu = u[i];
  float4 t;
  t.x = a.x + zz.x - uu.x;
  t.y = a.y + zz.y - uu.y;
  t.z = a.z + zz.z - uu.z;
  t.w = a.w + zz.w - uu.w;
  T[i] = t;
}

// dst[c][r] = src[r][c];  src: [R][Ccols]
__global__ void k_transpose(const float* __restrict__ src, float* __restrict__ dst,
                            int R, int Ccols) {
  __shared__ float t[32][33];
  const int tx = threadIdx.x & 31;
  const int ty = threadIdx.x >> 5;  // 0..7
  const int bx = blockIdx.x * 32;   // over Ccols
  const int by = blockIdx.y * 32;   // over R
#pragma unroll
  for (int i = 0; i < 4; ++i) {
    const int r = ty + i * 8;
    t[r][tx] = src[(size_t)(by + r) * Ccols + bx + tx];
  }
  __syncthreads();
#pragma unroll
  for (int i = 0; i < 4; ++i) {
    const int r = ty + i * 8;
    dst[(size_t)(bx + r) * R + by + tx] = t[tx][r];
  }
}

// ------------------------------- launcher ----------------------------------

extern "C" void kernel_launch(void* const* d_in, const int* in_sizes, int n_in,
                              void* d_out, int out_size, void* d_ws,
                              size_t ws_size, hipStream_t stream) {
  (void)in_sizes; (void)n_in; (void)out_size; (void)ws_size;

  constexpr int BATCH = 2048, IN = 1024, OUT = 2048;
  constexpr int MAX_ITERS = 100, NEWTON_ITERS = 8;
  constexpr float LAMBDA = 0.2f;
  constexpr float ALPHA = 0.25f;  // ~2/(lmin+lmax), spec(M) in [1, ~6.9]

  const float* x = (const float*)d_in[0];  // [2048][1024]
  const float* W = (const float*)d_in[1];  // [2048][1024] row-normalized

  const size_t mat = (size_t)OUT * OUT;  // 4M floats
  float* ws   = (float*)d_ws;            // needs 6*16MB + 8MB = 104 MB
  float* Mbuf = ws + 0 * mat;
  float* Xa   = ws + 1 * mat;
  float* Xb   = ws + 2 * mat;
  float* T    = ws + 3 * mat;
  float* Atb  = ws + 4 * mat;
  float* U    = ws + 5 * mat;
  float* Wt   = ws + 6 * mat;            // [1024][2048]
  float* z    = (float*)d_out;           // encoded = z: [2048][2048]
  float* dec  = (float*)d_out + mat;     // decoded: [2048][1024]

  const dim3 blk(128);
  const dim3 gSq(OUT / TB, OUT / TB);    // 16x16 blocks
  const dim3 gDec(IN / TB, BATCH / TB);  // 8x16 blocks
  const int eBlocks4 = (int)(mat / 4 / 256);

  // 0) Wt = W^T (for decode)
  k_transpose<<<dim3(IN / 32, OUT / 32), dim3(256), 0, stream>>>(W, Wt, OUT, IN);

  // 1) M = W @ W^T + I
  gemm_nt_wmma<EPI_ADD_IDENT><<<gSq, blk, 0, stream>>>(
      W, W, nullptr, Mbuf, nullptr, OUT, OUT, IN, 0.0f);

  // 2) Newton-Schulz: X0 = alpha*I; X <- 2X - X(MX). All iterates symmetric.
  k_init_identity<<<(int)(mat / 256), dim3(256), 0, stream>>>(Xa, OUT, ALPHA);
  float* cur = Xa;
  float* nxt = Xb;
  for (int it = 0; it < NEWTON_ITERS; ++it) {
    gemm_nt_wmma<EPI_NONE><<<gSq, blk, 0, stream>>>(
        Mbuf, cur, nullptr, T, nullptr, OUT, OUT, OUT, 0.0f);  // T = M@X (sym)
    gemm_nt_wmma<EPI_2E_MINUS><<<gSq, blk, 0, stream>>>(
        cur, T, cur, nxt, nullptr, OUT, OUT, OUT, 0.0f);       // 2X - X@T
    float* tmp = cur; cur = nxt; nxt = tmp;
  }
  // cur == M^-1 (symmetric)

  // 3) Atb = x @ W^T
  gemm_nt_wmma<EPI_NONE><<<gSq, blk, 0, stream>>>(
      x, W, nullptr, Atb, nullptr, BATCH, OUT, IN, 0.0f);

  // 4) z = 0, u = 0
  k_zero2<<<eBlocks4, dim3(256), 0, stream>>>((float4*)z, (float4*)U);

  // 5) 100 ADMM iterations (M^-1 symmetric -> NT)
  for (int it = 0; it < MAX_ITERS; ++it) {
    k_residual<<<eBlocks4, dim3(256), 0, stream>>>(
        (float4*)T, (const float4*)Atb, (const float4*)z, (const float4*)U);
    gemm_nt_wmma<EPI_ADMM><<<gSq, blk, 0, stream>>>(
        T, cur, nullptr, z, U, BATCH, OUT, OUT, LAMBDA);
  }

  // 6) decoded = z @ W = z @ (Wt)^T
  gemm_nt_wmma<EPI_NONE><<<gDec, blk, 0, stream>>>(
      z, Wt, nullptr, dec, nullptr, BATCH, IN, OUT, 0.0f);
}